// GE2ELoss_71571335021206
// MI455X (gfx1250) — compile-verified
//
#include <hip/hip_runtime.h>
#include <hip/hip_bf16.h>
#include <math.h>

// GE2E softmax loss, N=1024 speakers, M=16 utts, D=256 dims.
// Pipeline: k1 (centroids/norms/diag) -> k2 (WMMA f32 GEMM + fused logsumexp) -> k3 (reduce).

typedef __attribute__((ext_vector_type(2))) float v2f;
typedef __attribute__((ext_vector_type(8))) float v8f;

#define NSPK 1024
#define MUTT 16
#define DDIM 256
#define NROW (NSPK * MUTT)   // 16384
#define EPSV 1e-6f

// ---------------- workspace layout (in floats) ----------------
// centP  : [DDIM/2][NSPK][2]  transposed-pair centroids   = 262144
// invnc  : [NSPK]                                          =   1024
// invnd  : [NROW]                                          =  16384
// diagv  : [NROW]   (= w*cos_loo + b)                      =  16384
// part   : [256]    per-block partial losses               =    256
#define OFF_CENTP  0
#define OFF_INVNC  262144
#define OFF_INVND  263168
#define OFF_DIAG   279552
#define OFF_PART   295936

// ---------------------------------------------------------------
// block reduce (256 threads); result valid on thread 0 only
__device__ __forceinline__ float blockReduce256(float v, float* sh) {
    #pragma unroll
    for (int d = 1; d < 32; d <<= 1) v += __shfl_xor(v, d, 32);
    int w = threadIdx.x >> 5, l = threadIdx.x & 31;
    __syncthreads();                    // protect sh from previous use
    if (l == 0) sh[w] = v;
    __syncthreads();
    float t = 0.f;
    if (threadIdx.x == 0) {
        #pragma unroll
        for (int i = 0; i < 8; ++i) t += sh[i];
    }
    return t;
}

// Kernel 1: per-speaker centroids (transposed-pair layout), inv norms,
// and leave-one-out diagonal similarities.
__global__ __launch_bounds__(256) void ge2e_prep(
        const float* __restrict__ dv, const float* __restrict__ wp,
        const float* __restrict__ bp, float* __restrict__ ws) {
    __shared__ float sh[8];
    const int j = blockIdx.x;       // speaker
    const int d = threadIdx.x;      // dim (DDIM == blockDim.x == 256)

    float* centP = ws + OFF_CENTP;
    float* invnc = ws + OFF_INVNC;
    float* invnd = ws + OFF_INVND;
    float* diagv = ws + OFF_DIAG;

    const float* base = dv + (size_t)j * MUTT * DDIM;
    float x[MUTT];
    float s = 0.f;
    #pragma unroll
    for (int i = 0; i < MUTT; ++i) { x[i] = base[i * DDIM + d]; s += x[i]; }

    const float cent = s * (1.0f / (float)MUTT);
    // transposed-pair layout: centP[(d>>1)*2048 + j*2 + (d&1)]
    centP[(d >> 1) * (2 * NSPK) + j * 2 + (d & 1)] = cent;

    float nc2 = blockReduce256(cent * cent, sh);
    if (threadIdx.x == 0) invnc[j] = 1.0f / sqrtf(nc2);

    const float wv = wp[0], bv = bp[0];
    for (int i = 0; i < MUTT; ++i) {
        const float loo = (s - x[i]) * (1.0f / (float)(MUTT - 1));
        float pd = blockReduce256(x[i] * loo, sh);
        float pl = blockReduce256(loo * loo, sh);
        float pn = blockReduce256(x[i] * x[i], sh);
        if (threadIdx.x == 0) {
            const float nd = sqrtf(pn), nl = sqrtf(pl);
            const float cosloo = pd / fmaxf(nd * nl, EPSV);
            diagv[j * MUTT + i] = wv * cosloo + bv;
            invnd[j * MUTT + i] = 1.0f / nd;
        }
    }
}

// Kernel 2: fused GEMM (WMMA f32 16x16x4) + scaled-cos epilogue + diag
// substitution + shifted exp-sum; emits per-block partial loss.
__global__ __launch_bounds__(128) void ge2e_main(
        const float* __restrict__ dv, const float* __restrict__ ws_ro,
        const float* __restrict__ wp, const float* __restrict__ bp,
        float* __restrict__ part) {
    __shared__ float smem[4 * 16 * 256];     // 64 KB: 4 waves x (16 rows x 256 f32)

    const int wave = threadIdx.x >> 5;
    const int lane = threadIdx.x & 31;
    const int m    = lane & 15;              // row within tile / col within tile
    const int hi   = lane >> 4;              // half-wave select

    const int jspk    = blockIdx.x * 4 + wave;   // this wave's speaker
    const int rowBase = jspk * MUTT;             // 16 rows of this speaker

    const float* centP = ws_ro + OFF_CENTP;
    const float* invnc = ws_ro + OFF_INVNC;
    const float* invnd = ws_ro + OFF_INVND;
    const float* diagg = ws_ro + OFF_DIAG;

    const float wv = wp[0], bv = bp[0];
    const float shift = fabsf(wv) + bv;          // |cos|<=1 -> sim <= shift

    // ---- stage A tile (16x256 f32, contiguous in dvecs) into LDS, XOR-swizzled
    float* aw = smem + wave * 4096;
    const float4* src = (const float4*)(dv + (size_t)rowBase * DDIM);
    #pragma unroll
    for (int it = 0; it < 32; ++it) {
        const int t   = it * 32 + lane;          // float4 index 0..1023
        const int row = t >> 6;                  // 64 float4 per row
        const int c4  = t & 63;
        *(float4*)(aw + row * 256 + ((c4 ^ row) << 2)) = src[t];
    }
    __syncthreads();

    // ---- pull A into registers in WMMA 16x4 f32 layout:
    // lane(m,hi) holds A[m][4k+2hi], A[m][4k+2hi+1]
    v2f a[64];
    #pragma unroll
    for (int kk = 0; kk < 64; ++kk) {
        const int off = m * 256 + ((kk ^ m) << 2) + 2 * hi;   // 8B aligned
        a[kk] = *(const v2f*)(aw + off);
    }

    float invnd_r[8], diagv[8];
    #pragma unroll
    for (int r = 0; r < 8; ++r) {
        const int row = rowBase + r + 8 * hi;
        invnd_r[r] = invnd[row];
        diagv[r]   = diagg[row];
    }

    float sacc[8];
    #pragma unroll
    for (int r = 0; r < 8; ++r) sacc[r] = 0.f;

    const int ktDiag  = jspk >> 4;    // column tile containing k == jspk
    const int colDiag = jspk & 15;

    const v2f* bvec = (const v2f*)centP;   // pair (d, d+1) of cent[k] at [(d>>1)*1024 + k]

    for (int kt = 0; kt < 64; ++kt) {
        const int col = kt * 16 + m;
        const float sc_col = wv * invnc[col];

        v8f c0 = {}; v8f c1 = {};
        #pragma unroll
        for (int kk = 0; kk < 64; kk += 2) {
            // B lane(m=col,hi): cent[col][4kk+2hi], cent[col][4kk+2hi+1]
            v2f b0 = bvec[(2 * kk + hi) * NSPK + col];
            c0 = __builtin_amdgcn_wmma_f32_16x16x4_f32(
                    false, a[kk], false, b0, (short)0, c0, false, false);
            v2f b1 = bvec[(2 * (kk + 1) + hi) * NSPK + col];
            c1 = __builtin_amdgcn_wmma_f32_16x16x4_f32(
                    false, a[kk + 1], false, b1, (short)0, c1, false, false);
        }

        const bool isd = (kt == ktDiag) && (m == colDiag);
        #pragma unroll
        for (int r = 0; r < 8; ++r) {
            float v = (c0[r] + c1[r]) * (sc_col * invnd_r[r]) + bv;
            v = isd ? diagv[r] : v;          // leave-one-out diagonal
            sacc[r] += __expf(v - shift);    // shifted exp-sum (always <= 1)
        }
    }

    // ---- per-row logsumexp finish: sum each row's 16 lane-partials
    float lanesum = 0.f;
    #pragma unroll
    for (int r = 0; r < 8; ++r) {
        float S = sacc[r];
        S += __shfl_xor(S, 1, 32);
        S += __shfl_xor(S, 2, 32);
        S += __shfl_xor(S, 4, 32);
        S += __shfl_xor(S, 8, 32);           // butterfly within each 16-lane half
        lanesum += (shift + logf(S)) - diagv[r];   // lse - sim_diag
    }

    __syncthreads();                          // A tile no longer needed; reuse smem
    if (m == 0) smem[wave * 2 + hi] = lanesum;
    __syncthreads();
    if (threadIdx.x == 0) {
        float t = 0.f;
        #pragma unroll
        for (int i = 0; i < 8; ++i) t += smem[i];
        part[blockIdx.x] = t;
    }
}

// Kernel 3: deterministic reduction of 256 partials -> scalar loss.
__global__ __launch_bounds__(32) void ge2e_finish(
        const float* __restrict__ part, float* __restrict__ out) {
    const int lane = threadIdx.x;
    float s = 0.f;
    #pragma unroll
    for (int i = 0; i < 8; ++i) s += part[lane + 32 * i];
    #pragma unroll
    for (int d = 1; d < 32; d <<= 1) s += __shfl_xor(s, d, 32);
    if (lane == 0) out[0] = s;
}

extern "C" void kernel_launch(void* const* d_in, const int* in_sizes, int n_in,
                              void* d_out, int out_size, void* d_ws, size_t ws_size,
                              hipStream_t stream) {
    const float* dvecs = (const float*)d_in[0];   // [1024,16,256] f32
    const float* wp    = (const float*)d_in[1];   // scalar
    const float* bp    = (const float*)d_in[2];   // scalar
    float* ws  = (float*)d_ws;
    float* out = (float*)d_out;

    ge2e_prep<<<NSPK, 256, 0, stream>>>(dvecs, wp, bp, ws);
    ge2e_main<<<NSPK / 4, 128, 0, stream>>>(dvecs, ws, wp, bp, ws + OFF_PART);
    ge2e_finish<<<1, 32, 0, stream>>>(ws + OFF_PART, out);
}